// Model_24283745092197
// MI455X (gfx1250) — compile-verified
//
#include <hip/hip_runtime.h>
#include <math.h>

// Problem constants (match reference)
#define B_   16
#define N_   4096
#define C_   64
#define S_   1024      // GROUPS
#define K_   24        // kneighbors
#define D_   67        // C+3
#define D2_  131       // 2C+3
#define EPS_ 1e-5f

typedef float v2f __attribute__((ext_vector_type(2)));
typedef float v8f __attribute__((ext_vector_type(8)));

// ---------------------------------------------------------------------------
// Kernel 0: zero a float buffer (sq accumulator must be 0 every call)
// ---------------------------------------------------------------------------
__global__ void zero_kernel(float* __restrict__ p, int n) {
    int i = blockIdx.x * blockDim.x + threadIdx.x;
    if (i < n) p[i] = 0.0f;
}

// ---------------------------------------------------------------------------
// Kernel 1: furthest point sampling. One 1024-thread block per batch.
// xyz cached in LDS (48KB). dists live in registers (4 pts/thread).
// Writes fps_idx[B,S], qn[B,S] (=|q|^2), new_xyz[B,S,3] (start of d_out).
// ---------------------------------------------------------------------------
__global__ __launch_bounds__(1024) void fps_kernel(
    const float* __restrict__ xyz, int* __restrict__ fps_idx,
    float* __restrict__ qn, float* __restrict__ new_xyz)
{
    __shared__ float sx[N_], sy[N_], sz[N_];
    __shared__ float red_v[32];
    __shared__ int   red_i[32];
    __shared__ int   s_win;

    const int b    = blockIdx.x;
    const int tid  = threadIdx.x;
    const int lane = tid & 31;
    const int wave = tid >> 5;

    const float* base = xyz + (size_t)b * N_ * 3;
    for (int n = tid; n < N_; n += 1024) {
        sx[n] = base[n * 3 + 0];
        sy[n] = base[n * 3 + 1];
        sz[n] = base[n * 3 + 2];
    }
    __syncthreads();

    float dmin[4];
#pragma unroll
    for (int j = 0; j < 4; ++j) dmin[j] = 1e10f;

    int far = 0;
    for (int s = 0; s < S_; ++s) {
        if (tid == 0) {
            float cx = sx[far], cy = sy[far], cz = sz[far];
            fps_idx[b * S_ + s] = far;
            new_xyz[((size_t)b * S_ + s) * 3 + 0] = cx;
            new_xyz[((size_t)b * S_ + s) * 3 + 1] = cy;
            new_xyz[((size_t)b * S_ + s) * 3 + 2] = cz;
            qn[b * S_ + s] = cx * cx + cy * cy + cz * cz;
        }
        const float cx = sx[far], cy = sy[far], cz = sz[far];

        float bv = -1.0f; int bi = 0;
#pragma unroll
        for (int j = 0; j < 4; ++j) {
            int n = tid + j * 1024;
            float dx = sx[n] - cx, dy = sy[n] - cy, dz = sz[n] - cz;
            float d = dx * dx + dy * dy + dz * dz;
            dmin[j] = fminf(dmin[j], d);
            if (dmin[j] > bv || (dmin[j] == bv && n < bi)) { bv = dmin[j]; bi = n; }
        }
        // wave32 butterfly argmax (prefer lower index on ties, like jnp.argmax)
#pragma unroll
        for (int off = 16; off > 0; off >>= 1) {
            float ov = __shfl_xor(bv, off, 32);
            int   oi = __shfl_xor(bi, off, 32);
            if (ov > bv || (ov == bv && oi < bi)) { bv = ov; bi = oi; }
        }
        if (lane == 0) { red_v[wave] = bv; red_i[wave] = bi; }
        __syncthreads();
        if (wave == 0) {
            bv = red_v[lane]; bi = red_i[lane];
#pragma unroll
            for (int off = 16; off > 0; off >>= 1) {
                float ov = __shfl_xor(bv, off, 32);
                int   oi = __shfl_xor(bi, off, 32);
                if (ov > bv || (ov == bv && oi < bi)) { bv = ov; bi = oi; }
            }
            if (lane == 0) s_win = bi;
        }
        __syncthreads();
        far = s_win;
    }
}

// ---------------------------------------------------------------------------
// Kernel 2: squared-distance tiles via V_WMMA_F32_16X16X4_F32.
// D = (-2*A) x B + (qn[m]+pn[n])  ==  |q-p|^2, the reference's expanded form.
// A: 16 queries x 4 (xyz,0), B: 4 x 16 points. One wave per 16x16 tile.
// Per-batch dist buffer [S,N] in workspace.
// A layout (32b 16x4): lanes 0-15 hold K=0/1 in v0/v1; lanes 16-31 hold K=2/3.
// B layout (32b 4x16): v0 = rows K=0 (lanes0-15) / K=2 (lanes16-31); v1 = K=1/K=3.
// C/D layout: lanes 0-15 -> M=r, N=lane; lanes 16-31 -> M=r+8, N=lane-16.
// ---------------------------------------------------------------------------
__global__ __launch_bounds__(128) void dist_kernel(
    int b, const float* __restrict__ xyz, const float* __restrict__ new_xyz,
    const float* __restrict__ qn, float* __restrict__ dist)
{
    const int lane = threadIdx.x & 31;
    const int wave = threadIdx.x >> 5;
    const int s0 = blockIdx.x * 16;
    const int n0 = (blockIdx.y * 4 + wave) * 16;
    const int hi = lane >> 4;    // 0: lanes 0-15, 1: lanes 16-31
    const int l  = lane & 15;

    // A matrix (queries), pre-scaled by -2
    const float* q = new_xyz + ((size_t)b * S_ + s0 + l) * 3;
    float qx = q[0], qy = q[1], qz = q[2];
    v2f A;
    A.x = hi ? (-2.0f * qz) : (-2.0f * qx);
    A.y = hi ? 0.0f         : (-2.0f * qy);

    // B matrix (points) + local point norm
    const float* p = xyz + ((size_t)b * N_ + n0 + l) * 3;
    float px = p[0], py = p[1], pz = p[2];
    v2f Bm;
    Bm.x = hi ? pz : px;
    Bm.y = hi ? 0.0f : py;
    float pn = px * px + py * py + pz * pz;

    // C = qn[m] + pn[n]
    v8f Cc;
#pragma unroll
    for (int r = 0; r < 8; ++r)
        Cc[r] = qn[b * S_ + s0 + r + 8 * hi] + pn;

    // (neg_a, A, neg_b, B, c_mod, C, reuse_a, reuse_b)
    v8f Dm = __builtin_amdgcn_wmma_f32_16x16x4_f32(
        false, A, false, Bm, (short)0, Cc, false, false);

    float* drow = dist + (size_t)s0 * N_ + n0 + l;
#pragma unroll
    for (int r = 0; r < 8; ++r)
        drow[(size_t)(r + 8 * hi) * N_] = Dm[r];
}

// ---------------------------------------------------------------------------
// Kernel 3: top-k (k=24 smallest) per query row. One wave per row.
// Each lane insertion-sorts a 128-candidate slice into a register top-24
// (fully unrolled, no dynamic indexing), then 32-way merge via LDS + shuffles.
// Ties broken by lower point index (matches lax.top_k on -sqr).
// ---------------------------------------------------------------------------
__global__ __launch_bounds__(32) void topk_kernel(
    int b, const float* __restrict__ dist, int* __restrict__ knn)
{
    const int s = blockIdx.x;
    const int lane = threadIdx.x;
    const float* row = dist + (size_t)s * N_;

    float ld[K_]; int li[K_];
#pragma unroll
    for (int p = 0; p < K_; ++p) { ld[p] = 3.0e38f; li[p] = 0x7fffffff; }

    for (int j = 0; j < N_ / 32; ++j) {
        int n = lane + j * 32;
        float v = row[n];
        if (v < ld[K_ - 1] || (v == ld[K_ - 1] && n < li[K_ - 1])) {
            ld[K_ - 1] = v; li[K_ - 1] = n;
#pragma unroll
            for (int p = K_ - 1; p > 0; --p) {
                bool sw = (ld[p] < ld[p - 1]) ||
                          (ld[p] == ld[p - 1] && li[p] < li[p - 1]);
                if (sw) {
                    float tv = ld[p]; ld[p] = ld[p - 1]; ld[p - 1] = tv;
                    int   ti = li[p]; li[p] = li[p - 1]; li[p - 1] = ti;
                }
            }
        }
    }

    __shared__ float sv[32 * K_];
    __shared__ int   si[32 * K_];
#pragma unroll
    for (int p = 0; p < K_; ++p) { sv[lane * K_ + p] = ld[p]; si[lane * K_ + p] = li[p]; }
    __syncthreads();

    int pos = 0;
    for (int r = 0; r < K_; ++r) {
        float v = (pos < K_) ? sv[lane * K_ + pos] : 3.0e38f;
        int   i = (pos < K_) ? si[lane * K_ + pos] : 0x7fffffff;
        int   wl = lane;
#pragma unroll
        for (int off = 16; off > 0; off >>= 1) {
            float ov = __shfl_xor(v,  off, 32);
            int   oi = __shfl_xor(i,  off, 32);
            int   ow = __shfl_xor(wl, off, 32);
            if (ov < v || (ov == v && oi < i)) { v = ov; i = oi; wl = ow; }
        }
        if (lane == 0) knn[((size_t)b * S_ + s) * K_ + r] = i;
        if (lane == wl) ++pos;
    }
}

// ---------------------------------------------------------------------------
// Kernel 4: per-(b,s) group mean over k, plus global sum of diff^2 per (b,d).
// grouped feature d: d<64 -> points[..,d]; d in 64..66 -> xyz coord d-64.
// ---------------------------------------------------------------------------
__global__ __launch_bounds__(96) void stat_kernel(
    const float* __restrict__ xyz, const float* __restrict__ points,
    const int* __restrict__ knn, float* __restrict__ mean, float* __restrict__ sq)
{
    const int bs = blockIdx.x;
    const int b = bs >> 10;
    const int t = threadIdx.x;

    __shared__ int nidx[K_];
    if (t < K_) nidx[t] = knn[(size_t)bs * K_ + t];
    __syncthreads();

    if (t < D_) {
        float v[K_];
        float sum = 0.0f;
#pragma unroll
        for (int kk = 0; kk < K_; ++kk) {
            int n = nidx[kk];
            float x = (t < C_) ? points[((size_t)b * N_ + n) * C_ + t]
                               : xyz[((size_t)b * N_ + n) * 3 + (t - C_)];
            v[kk] = x; sum += x;
        }
        float m = sum * (1.0f / K_);
        mean[(size_t)bs * D_ + t] = m;
        float ss = 0.0f;
#pragma unroll
        for (int kk = 0; kk < K_; ++kk) { float d = v[kk] - m; ss += d * d; }
        atomicAdd(&sq[b * D_ + t], ss);
    }
}

// std = sqrt(sum(diff^2) / (S*k - 1))   (diff mean over S*k is exactly 0)
__global__ void std_kernel(const float* __restrict__ sq, float* __restrict__ stdv, int n)
{
    int i = blockIdx.x * blockDim.x + threadIdx.x;
    if (i < n) stdv[i] = sqrtf(sq[i] / (float)(S_ * K_ - 1));
}

// ---------------------------------------------------------------------------
// Kernel 5: final write: out[b,s,kk,d] for d<67: alpha*(x-mean)/(std+eps)+beta,
// d>=67: replicated center features (points at fps_idx). points is L2-resident.
// ---------------------------------------------------------------------------
__global__ __launch_bounds__(128) void write_kernel(
    const float* __restrict__ xyz, const float* __restrict__ points,
    const int* __restrict__ knn, const int* __restrict__ fps_idx,
    const float* __restrict__ mean, const float* __restrict__ stdv,
    const float* __restrict__ alpha, const float* __restrict__ beta,
    float* __restrict__ out)
{
    const int bs = blockIdx.x;
    const int b = bs >> 10;
    const int t = threadIdx.x;

    __shared__ int   nidx[K_];
    __shared__ float smean[D_], sscale[D_], sbeta[D_], srep[C_];

    if (t < K_) nidx[t] = knn[(size_t)bs * K_ + t];
    if (t < D_) {
        smean[t]  = mean[(size_t)bs * D_ + t];
        sscale[t] = alpha[t] / (stdv[b * D_ + t] + EPS_);
        sbeta[t]  = beta[t];
    }
    if (t < C_) {
        int c0 = fps_idx[bs];
        srep[t] = points[((size_t)b * N_ + c0) * C_ + t];
    }
    __syncthreads();

    float* obase = out + (size_t)bs * K_ * D2_;
    for (int kk = 0; kk < K_; ++kk) {
        int n = nidx[kk];
        for (int d = t; d < D2_; d += 128) {
            float o;
            if (d < D_) {
                float x = (d < C_) ? points[((size_t)b * N_ + n) * C_ + d]
                                   : xyz[((size_t)b * N_ + n) * 3 + (d - C_)];
                o = (x - smean[d]) * sscale[d] + sbeta[d];
            } else {
                o = srep[d - D_];
            }
            obase[(size_t)kk * D2_ + d] = o;
        }
    }
}

// ---------------------------------------------------------------------------
extern "C" void kernel_launch(void* const* d_in, const int* in_sizes, int n_in,
                              void* d_out, int out_size, void* d_ws, size_t ws_size,
                              hipStream_t stream)
{
    (void)in_sizes; (void)n_in; (void)out_size; (void)ws_size;
    const float* xyz    = (const float*)d_in[0];
    const float* points = (const float*)d_in[1];
    const float* alpha  = (const float*)d_in[2];
    const float* beta   = (const float*)d_in[3];
    float* out = (float*)d_out;

    char* ws = (char*)d_ws;
    size_t off = 0;
    auto alloc = [&](size_t bytes) {
        size_t o = off;
        off = (off + bytes + 255) & ~(size_t)255;
        return o;
    };
    int*   fps_idx = (int*)  (ws + alloc((size_t)B_ * S_ * 4));
    float* qn      = (float*)(ws + alloc((size_t)B_ * S_ * 4));
    int*   knn     = (int*)  (ws + alloc((size_t)B_ * S_ * K_ * 4));
    float* mean    = (float*)(ws + alloc((size_t)B_ * S_ * D_ * 4));
    float* sq      = (float*)(ws + alloc((size_t)B_ * D_ * 4));
    float* stdv    = (float*)(ws + alloc((size_t)B_ * D_ * 4));
    float* dist    = (float*)(ws + alloc((size_t)S_ * N_ * 4));   // per-batch, reused

    float* new_xyz  = out;                          // [B,S,3]
    float* out_main = out + (size_t)B_ * S_ * 3;    // [B,S,K,131]

    zero_kernel<<<dim3((B_ * D_ + 127) / 128), dim3(128), 0, stream>>>(sq, B_ * D_);
    fps_kernel<<<dim3(B_), dim3(1024), 0, stream>>>(xyz, fps_idx, qn, new_xyz);
    for (int b = 0; b < B_; ++b) {
        dist_kernel<<<dim3(S_ / 16, N_ / 64), dim3(128), 0, stream>>>(b, xyz, new_xyz, qn, dist);
        topk_kernel<<<dim3(S_), dim3(32), 0, stream>>>(b, dist, knn);
    }
    stat_kernel<<<dim3(B_ * S_), dim3(96), 0, stream>>>(xyz, points, knn, mean, sq);
    std_kernel<<<dim3((B_ * D_ + 127) / 128), dim3(128), 0, stream>>>(sq, stdv, B_ * D_);
    write_kernel<<<dim3(B_ * S_), dim3(128), 0, stream>>>(
        xyz, points, knn, fps_idx, mean, stdv, alpha, beta, out_main);
}